// MeanBinningLayer_69312182223481
// MI455X (gfx1250) — compile-verified
//
#include <hip/hip_runtime.h>
#include <hip/hip_bf16.h>

typedef __attribute__((ext_vector_type(2))) float v2f;
typedef __attribute__((ext_vector_type(4))) float v4f;
typedef __attribute__((ext_vector_type(8))) float v8f;

#define DIM1          8192
#define DIM2          256
#define NBINS         64
#define ROWS_PER_BIN  128                       // DIM1 / NBINS
#define CHUNK_ROWS    16
#define NCHUNKS       (ROWS_PER_BIN / CHUNK_ROWS)  // 8
#define LDS_STRIDE    264                       // 256 + 8 dword pad -> conflict-free reads
#define THREADS       512

// One workgroup per (batch, bin) tile: 128 rows x 256 cols of f32.
// Pipeline: global_load_b128 (NT) -> LDS (double buffer) -> ds_load_b32 in
// WMMA striping -> V_WMMA_F32_16X16X4_F32 with A = ones (D rows = column sums).
__global__ __launch_bounds__(THREADS) void MeanBinning_kernel(
    const float* __restrict__ in, float* __restrict__ out) {
  __shared__ __align__(16) float lds[2][CHUNK_ROWS * LDS_STRIDE];

  const int blk   = blockIdx.x;        // 0 .. 32*64-1
  const int batch = blk >> 6;
  const int bin   = blk & 63;

  const int tid    = threadIdx.x;
  const int lane   = tid & 31;                   // wave32
  const int wave   = tid >> 5;                   // 0..15
  const int col    = (wave << 4) | (lane & 15);  // WMMA N = lane%16
  const int rowSel = (lane >> 4) << 1;           // lanes 0-15 -> rows {0,1}; 16-31 -> {2,3}

  // Stage-in mapping: thread t loads float4 at (row tid>>6, colgroup tid&63)
  // and (row (tid>>6)+8, same colgroup). Per-wave: 512B contiguous.
  const int srowA = tid >> 6;          // 0..7
  const int srowB = srowA + 8;         // 8..15
  const int scg   = tid & 63;          // float4 column group

  const float* gbase =
      in + ((size_t)batch * DIM1 + (size_t)bin * ROWS_PER_BIN) * DIM2;

  v2f ones;
  ones.x = 1.0f;
  ones.y = 1.0f;
  v8f acc = {};

  // Prologue: fetch + stage chunk 0 into buffer 0.
  v4f rA = __builtin_nontemporal_load(
      (const v4f*)(gbase + (size_t)srowA * DIM2 + (size_t)scg * 4));
  v4f rB = __builtin_nontemporal_load(
      (const v4f*)(gbase + (size_t)srowB * DIM2 + (size_t)scg * 4));
  *(v4f*)&lds[0][srowA * LDS_STRIDE + scg * 4] = rA;
  *(v4f*)&lds[0][srowB * LDS_STRIDE + scg * 4] = rB;
  __syncthreads();

  for (int k = 0; k < NCHUNKS; ++k) {
    const int cur = k & 1;

    // Issue next chunk's global loads early (latency hides under WMMA phase).
    if (k + 1 < NCHUNKS) {
      const float* cbase = gbase + (size_t)(k + 1) * CHUNK_ROWS * DIM2;
      rA = __builtin_nontemporal_load(
          (const v4f*)(cbase + (size_t)srowA * DIM2 + (size_t)scg * 4));
      rB = __builtin_nontemporal_load(
          (const v4f*)(cbase + (size_t)srowB * DIM2 + (size_t)scg * 4));
    }

    // Reduce this chunk: 4 quads of 4 rows -> 4 accumulating WMMAs.
    // Bank math: half-0 reads banks [s..s+15], half-1 [s+16..s+31] (2*264 % 64 == 16).
#pragma unroll
    for (int q = 0; q < CHUNK_ROWS / 4; ++q) {
      const int r0 = 4 * q + rowSel;
      v2f bmat;
      bmat.x = lds[cur][r0 * LDS_STRIDE + col];
      bmat.y = lds[cur][(r0 + 1) * LDS_STRIDE + col];
      acc = __builtin_amdgcn_wmma_f32_16x16x4_f32(
          /*neg_a=*/false, ones, /*neg_b=*/false, bmat,
          /*c_mod=*/(short)0, acc, /*reuse_a=*/false, /*reuse_b=*/false);
    }

    // Ping-pong: sync #1 = everyone done reading buffer !cur (read in k-1),
    // store chunk k+1 there, sync #2 = stores visible before next reads.
    if (k + 1 < NCHUNKS) {
      __syncthreads();
      const int nxt = cur ^ 1;
      *(v4f*)&lds[nxt][srowA * LDS_STRIDE + scg * 4] = rA;
      *(v4f*)&lds[nxt][srowB * LDS_STRIDE + scg * 4] = rB;
      __syncthreads();
    }
  }

  // All 16 D rows are identical column sums; lanes 0-15 hold N=0..15 in acc[0].
  if (lane < 16) {
    const float denom = 128.0f + 1e-8f;   // == 128.0f in fp32, faithful to reference
    const float scale = 1.0f / denom;
    out[((size_t)batch * NBINS + (size_t)bin) * DIM2 + (size_t)col] = acc[0] * scale;
  }
}

extern "C" void kernel_launch(void* const* d_in, const int* in_sizes, int n_in,
                              void* d_out, int out_size, void* d_ws, size_t ws_size,
                              hipStream_t stream) {
  (void)in_sizes; (void)n_in; (void)d_ws; (void)ws_size; (void)out_size;
  const float* in = (const float*)d_in[0];   // [32, 8192, 256] fp32
  float* out = (float*)d_out;                // [32, 64, 256] fp32

  const int batches = 32;
  dim3 grid(batches * NBINS);   // 2048 workgroups, one per (batch, bin)
  dim3 block(THREADS);          // 16 wave32 waves -> 256 columns
  MeanBinning_kernel<<<grid, block, 0, stream>>>(in, out);
}